// RsageLayer_5729486373122
// MI455X (gfx1250) — compile-verified
//
#include <hip/hip_runtime.h>
#include <hip/hip_bf16.h>

#define N_USER 50000
#define N_ITEM 50000
#define D 128
#define LDS_STRIDE 132   // 132 mod 64 == 4 -> conflict-free ds_load_b64 across 32 lanes

typedef float v2f __attribute__((ext_vector_type(2)));
typedef float v8f __attribute__((ext_vector_type(8)));

// ---------------------------------------------------------------- utilities
__global__ void zero_f32(float* __restrict__ p, size_t n) {
    size_t i = (size_t)blockIdx.x * blockDim.x + threadIdx.x;
    size_t stride = (size_t)gridDim.x * blockDim.x;
    for (; i < n; i += stride) p[i] = 0.0f;
}

// Wself_user = Wself_uu + Wself_iu ; b_user = b_uu + b_iu  (both dst-type == user)
__global__ void combine_user_w(const float* __restrict__ wa, const float* __restrict__ wb,
                               const float* __restrict__ ba, const float* __restrict__ bb,
                               float* __restrict__ wc, float* __restrict__ bc) {
    int i = blockIdx.x * blockDim.x + threadIdx.x;
    if (i < D * D) wc[i] = wa[i] + wb[i];
    if (i < D)     bc[i] = ba[i] + bb[i];
}

// ------------------------------------------------------- sparse scatter phase
// One wave32 per edge: lanes cooperatively move one 128-float source row into
// the destination accumulator with float atomics; lane 0 counts the degree.
__global__ __launch_bounds__(256) void scatter_mean_accum(
    const float* __restrict__ hsrc, const int* __restrict__ src,
    const int* __restrict__ dst, float* __restrict__ neigh,
    float* __restrict__ deg, int ne) {
    int wid  = (int)(((size_t)blockIdx.x * blockDim.x + threadIdx.x) >> 5);
    int lane = threadIdx.x & 31;
    if (wid >= ne) return;
    int s = src[wid];
    int d = dst[wid];
    const float4 v = *(const float4*)(hsrc + (size_t)s * D + lane * 4);
    float* p = neigh + (size_t)d * D + lane * 4;
    atomicAdd(p + 0, v.x);
    atomicAdd(p + 1, v.y);
    atomicAdd(p + 2, v.z);
    atomicAdd(p + 3, v.w);
    if (lane == 0) atomicAdd(deg + d, 1.0f);
}

// ---------------------------------------------------------- dense WMMA phase
// K=128 accumulation with V_WMMA_F32_16X16X4_F32.
// sA already offset by (lane&15)*LDS_STRIDE + koff ; wcol = W + koff*D + col.
__device__ __forceinline__ v8f tile_mm_f32(const float* __restrict__ sA,
                                           const float* __restrict__ wcol, v8f c) {
#pragma unroll
    for (int k = 0; k < D; k += 4) {
        v2f a = *(const v2f*)(sA + k);            // ds_load_b64, bank-conflict free
        v2f b;
        b.x = wcol[k * D];                        // B 4x16: lanes 0-15 K{0,1}, 16-31 K{2,3}
        b.y = wcol[k * D + D];
        c = __builtin_amdgcn_wmma_f32_16x16x4_f32(false, a, false, b, (short)0, c,
                                                  false, false);
    }
    return c;
}

// One block per 16-row M-tile; 8 waves cover the 8 N-tiles of D_OUT=128.
// out = A0 @ W0 + diag(1/max(deg1,1)) A1 @ W1 [+ diag(1/max(deg2,1)) A2 @ W2] + bias
__global__ __launch_bounds__(256) void sage_gemm_wmma(
    const float* __restrict__ A0, const float* __restrict__ W0,
    const float* __restrict__ A1, const float* __restrict__ W1,
    const float* __restrict__ deg1,
    const float* __restrict__ A2, const float* __restrict__ W2,
    const float* __restrict__ deg2,
    const float* __restrict__ bias, float* __restrict__ out) {
    __shared__ float lds[3 * 16 * LDS_STRIDE];
    const int tid  = threadIdx.x;
    const int row0 = blockIdx.x * 16;

    // Cooperative staging: 256 threads x 8 floats = one 16x128 tile per pass.
    // Degree normalization (mean aggregation) is fused into the A1/A2 stage.
    {
        int r  = tid >> 4;          // 0..15 tile row
        int c0 = (tid & 15) * 8;    // 0..120 column chunk
        const float* a0 = A0 + (size_t)(row0 + r) * D + c0;
        float* s0 = lds + r * LDS_STRIDE + c0;
#pragma unroll
        for (int t = 0; t < 8; ++t) s0[t] = a0[t];

        float inv1 = 1.0f / fmaxf(deg1[row0 + r], 1.0f);
        const float* a1 = A1 + (size_t)(row0 + r) * D + c0;
        float* s1 = lds + (16 * LDS_STRIDE) + r * LDS_STRIDE + c0;
#pragma unroll
        for (int t = 0; t < 8; ++t) s1[t] = a1[t] * inv1;

        if (A2) {
            float inv2 = 1.0f / fmaxf(deg2[row0 + r], 1.0f);
            const float* a2 = A2 + (size_t)(row0 + r) * D + c0;
            float* s2 = lds + (32 * LDS_STRIDE) + r * LDS_STRIDE + c0;
#pragma unroll
            for (int t = 0; t < 8; ++t) s2[t] = a2[t] * inv2;
        }
    }
    __syncthreads();

    const int lane = tid & 31;
    const int wave = tid >> 5;      // ntile 0..7
    const int l16  = lane & 15;
    const int half = lane >> 4;     // A/B: lanes 0-15 hold K{0,1}, lanes 16-31 K{2,3}
    const int koff = half * 2;
    const int col  = wave * 16 + l16;

    v8f c;
    float bv = bias[col];           // C(m,n) init = bias[n] for every m
#pragma unroll
    for (int j = 0; j < 8; ++j) c[j] = bv;

    const float* sA = lds + l16 * LDS_STRIDE + koff;
    c = tile_mm_f32(sA,                   W0 + koff * D + col, c);
    c = tile_mm_f32(sA + 16 * LDS_STRIDE, W1 + koff * D + col, c);
    if (A2)
        c = tile_mm_f32(sA + 32 * LDS_STRIDE, W2 + koff * D + col, c);

    // D layout: VGPR j -> row (half*8 + j), col = l16 within the N-tile.
    float* orow = out + (size_t)(row0 + half * 8) * D + col;
#pragma unroll
    for (int j = 0; j < 8; ++j) orow[(size_t)j * D] = c[j];
}

// ------------------------------------------------------------------- launch
extern "C" void kernel_launch(void* const* d_in, const int* in_sizes, int n_in,
                              void* d_out, int out_size, void* d_ws, size_t ws_size,
                              hipStream_t stream) {
    const float* h_user    = (const float*)d_in[0];
    const float* h_item    = (const float*)d_in[1];
    const float* Wself_uu  = (const float*)d_in[2];
    const float* Wneigh_uu = (const float*)d_in[3];
    const float* b_uu      = (const float*)d_in[4];
    const float* Wself_ui  = (const float*)d_in[5];
    const float* Wneigh_ui = (const float*)d_in[6];
    const float* b_ui      = (const float*)d_in[7];
    const float* Wself_iu  = (const float*)d_in[8];
    const float* Wneigh_iu = (const float*)d_in[9];
    const float* b_iu      = (const float*)d_in[10];
    const int* e_uu_src = (const int*)d_in[11];
    const int* e_uu_dst = (const int*)d_in[12];
    const int* e_ui_src = (const int*)d_in[13];
    const int* e_ui_dst = (const int*)d_in[14];
    const int* e_iu_src = (const int*)d_in[15];
    const int* e_iu_dst = (const int*)d_in[16];
    const int Euu = in_sizes[11], Eui = in_sizes[13], Eiu = in_sizes[15];

    // Workspace layout (floats)
    float* ws = (float*)d_ws;
    size_t o = 0;
    float* neigh_uu = ws + o; o += (size_t)N_USER * D;
    float* neigh_iu = ws + o; o += (size_t)N_USER * D;
    float* neigh_ui = ws + o; o += (size_t)N_ITEM * D;
    float* deg_uu   = ws + o; o += N_USER;
    float* deg_iu   = ws + o; o += N_USER;
    float* deg_ui   = ws + o; o += N_ITEM;
    float* Wself_u  = ws + o; o += D * D;
    float* b_user   = ws + o; o += D;

    zero_f32<<<2048, 256, 0, stream>>>(ws, o);
    combine_user_w<<<(D * D + 255) / 256, 256, 0, stream>>>(Wself_uu, Wself_iu,
                                                            b_uu, b_iu, Wself_u, b_user);

    scatter_mean_accum<<<(Euu + 7) / 8, 256, 0, stream>>>(h_user, e_uu_src, e_uu_dst,
                                                          neigh_uu, deg_uu, Euu);
    scatter_mean_accum<<<(Eiu + 7) / 8, 256, 0, stream>>>(h_item, e_iu_src, e_iu_dst,
                                                          neigh_iu, deg_iu, Eiu);
    scatter_mean_accum<<<(Eui + 7) / 8, 256, 0, stream>>>(h_user, e_ui_src, e_ui_dst,
                                                          neigh_ui, deg_ui, Eui);

    float* out_user = (float*)d_out;
    float* out_item = out_user + (size_t)N_USER * D;

    sage_gemm_wmma<<<N_USER / 16, 256, 0, stream>>>(
        h_user, Wself_u, neigh_uu, Wneigh_uu, deg_uu,
        neigh_iu, Wneigh_iu, deg_iu, b_user, out_user);
    sage_gemm_wmma<<<N_ITEM / 16, 256, 0, stream>>>(
        h_item, Wself_ui, neigh_ui, Wneigh_ui, deg_ui,
        nullptr, nullptr, nullptr, b_ui, out_item);
}